// ProtoHead_44873818308722
// MI455X (gfx1250) — compile-verified
//
#include <hip/hip_runtime.h>

// ProtoHead masked squared-Euclidean logits via fp32 WMMA on gfx1250.
//
// logits[b,n,m] = scale/D * ( sum_d q^2*mask^2 - 2*sum_d q*(mask^2*p) + sum_d (mask*p)^2 )
//
// One fused WMMA loop (two K=1024 GEMM streams sharing accumulators);
// B operands (mask^2, -2*mask^2*p) derived once at staging time; p_term via
// LDS float atomics; query tile staged with async-to-LDS when available.

typedef __attribute__((ext_vector_type(2))) float v2f;
typedef __attribute__((ext_vector_type(8))) float v8f;
typedef __attribute__((ext_vector_type(4))) int   v4i;

#define BS_   8
#define NQ_   2048
#define MP_   64
#define DD_   1024
#define BK_   32
#define BN_   128
#define QSTR_ (BK_ + 4)   // LDS row stride (floats): float4-aligned, breaks bank repeats

#if defined(__has_builtin)
#if __has_builtin(__builtin_amdgcn_global_load_async_to_lds_b128)
#define HAVE_ASYNC_LDS 1
#endif
#if __has_builtin(__builtin_amdgcn_s_wait_asynccnt)
#define HAVE_WAIT_ASYNC_BUILTIN 1
#endif
#endif

#if defined(HAVE_ASYNC_LDS)
// Parameter types per hipcc diagnostic: arg0 is 'int __vector(4) __device__ *'
// (v4i in address_space(1), non-const); arg1 assumed v4i in address_space(3).
typedef __attribute__((address_space(1))) v4i* gv4i_p;
typedef __attribute__((address_space(3))) v4i* lv4i_p;

static __device__ __forceinline__ void wait_asynccnt0() {
#if defined(HAVE_WAIT_ASYNC_BUILTIN)
    __builtin_amdgcn_s_wait_asynccnt(0);
#else
    asm volatile("s_wait_asynccnt 0" ::: "memory");
#endif
}

static __device__ __forceinline__ void async_copy_b128(const float* gsrc, float* lds_dst) {
    float* gnc = const_cast<float*>(gsrc);
    __builtin_amdgcn_global_load_async_to_lds_b128(
        (gv4i_p)gnc, (lv4i_p)lds_dst, 0, 0);
}
#endif

__global__ __launch_bounds__(256)
void protohead_wmma_kernel(const float* __restrict__ proto,
                           const float* __restrict__ mask,
                           const float* __restrict__ query,
                           const float* __restrict__ scale,
                           float* __restrict__ out)
{
    __shared__ float sQ  [BN_][QSTR_];   // raw query tile
    __shared__ float sBm2[MP_][QSTR_];   // mask^2
    __shared__ float sBmp[MP_][QSTR_];   // -2*mask^2*p
    __shared__ float sPterm[MP_];

    const int b    = blockIdx.y;
    const int n0   = blockIdx.x * BN_;
    const int tid  = threadIdx.x;
    const int wave = tid >> 5;
    const int lane = tid & 31;
    const int l16  = lane & 15;
    const int half = lane >> 4;          // 0: lanes 0-15, 1: lanes 16-31

    if (tid < MP_) sPterm[tid] = 0.0f;

    v8f acc[4];
    const v8f vzero = {0.f, 0.f, 0.f, 0.f, 0.f, 0.f, 0.f, 0.f};
#pragma unroll
    for (int mt = 0; mt < 4; ++mt) acc[mt] = vzero;

    const float* qbase = query + ((size_t)b * NQ_ + n0) * DD_;
    const float* mbase = mask  + ((size_t)b * MP_) * DD_;
    const float* pbase = proto + ((size_t)b * MP_) * DD_;

    __syncthreads();

    const int arow = wave * 16 + l16;    // A-matrix row handled by this lane

    for (int k0 = 0; k0 < DD_; k0 += BK_) {
        // ---- stage query tile: BN_ x BK_ ----
        {
            const int r = tid >> 3;              // 0..31
            const int c = (tid & 7) << 2;        // 0,4,..,28
#pragma unroll
            for (int i = 0; i < 4; ++i) {
                const int row = r + i * 32;      // 0..127
#if defined(HAVE_ASYNC_LDS)
                async_copy_b128(qbase + (size_t)row * DD_ + k0 + c, &sQ[row][c]);
#else
                const float4 v = *(const float4*)(qbase + (size_t)row * DD_ + k0 + c);
                sQ[row][c + 0] = v.x;
                sQ[row][c + 1] = v.y;
                sQ[row][c + 2] = v.z;
                sQ[row][c + 3] = v.w;
#endif
            }
        }
        // ---- stage derived B tiles: MP_ x BK_, accumulate p_term ----
        {
#pragma unroll
            for (int i = 0; i < 2; ++i) {
                const int idx = tid + i * 256;   // 0..511
                const int m   = idx >> 3;        // 0..63
                const int c   = (idx & 7) << 2;
                const size_t off = (size_t)m * DD_ + k0 + c;
                const float4 mk = *(const float4*)(mbase + off);
                const float4 pr = *(const float4*)(pbase + off);
                // mask^2
                const float m2x = mk.x * mk.x, m2y = mk.y * mk.y;
                const float m2z = mk.z * mk.z, m2w = mk.w * mk.w;
                sBm2[m][c + 0] = m2x;  sBm2[m][c + 1] = m2y;
                sBm2[m][c + 2] = m2z;  sBm2[m][c + 3] = m2w;
                // -2*mask^2*p
                sBmp[m][c + 0] = -2.0f * m2x * pr.x;
                sBmp[m][c + 1] = -2.0f * m2y * pr.y;
                sBmp[m][c + 2] = -2.0f * m2z * pr.z;
                sBmp[m][c + 3] = -2.0f * m2w * pr.w;
                // p_term partial: (mask*p)^2
                const float a0 = mk.x * pr.x, a1 = mk.y * pr.y;
                const float a2 = mk.z * pr.z, a3 = mk.w * pr.w;
                atomicAdd(&sPterm[m], a0 * a0 + a1 * a1 + a2 * a2 + a3 * a3);
            }
        }
#if defined(HAVE_ASYNC_LDS)
        wait_asynccnt0();
#endif
        __syncthreads();

        // ---- WMMA over this K-chunk: 8 steps of K=4 ----
#pragma unroll
        for (int ks = 0; ks < BK_; ks += 4) {
            const int ka = ks + 2 * half;        // this lane's K pair base
            v2f aq;
            aq.x = sQ[arow][ka];
            aq.y = sQ[arow][ka + 1];
            const v2f aq2 = aq * aq;             // A-fragment for q^2 term
#pragma unroll
            for (int mt = 0; mt < 4; ++mt) {
                const int mrow = mt * 16 + l16;  // B column (= prototype index)
                v2f bm2, bmp;
                bm2.x = sBm2[mrow][ka];
                bm2.y = sBm2[mrow][ka + 1];
                bmp.x = sBmp[mrow][ka];
                bmp.y = sBmp[mrow][ka + 1];
                acc[mt] = __builtin_amdgcn_wmma_f32_16x16x4_f32(
                    false, aq2, false, bm2, (short)0, acc[mt], false, false);
                acc[mt] = __builtin_amdgcn_wmma_f32_16x16x4_f32(
                    false, aq,  false, bmp, (short)0, acc[mt], false, false);
            }
        }
        __syncthreads();
    }

    // p_term atomics all landed before the final compute barrier.
    const float sc = scale[0] * (1.0f / (float)DD_);
#pragma unroll
    for (int mt = 0; mt < 4; ++mt) {
        const int col = mt * 16 + l16;
        const float pv = sPterm[col];
#pragma unroll
        for (int v = 0; v < 8; ++v) {
            // C/D layout: VGPR v -> M = v + 8*half, lane%16 -> N
            const int row = n0 + wave * 16 + v + 8 * half;
            out[((size_t)b * NQ_ + row) * MP_ + col] = sc * (acc[mt][v] + pv);
        }
    }
}

extern "C" void kernel_launch(void* const* d_in, const int* in_sizes, int n_in,
                              void* d_out, int out_size, void* d_ws, size_t ws_size,
                              hipStream_t stream) {
    // setup_inputs order:
    // 0: prototypes [8,64,1024] f32
    // 1: masktypes  [8,64,1024] f32
    // 2: query      [8,2048,1024] f32
    // 3: support        (unused)
    // 4: support_labels (unused)
    // 5: scale      [1] f32
    // 6: n_way (unused), 7: n_shot (unused)
    const float* proto = (const float*)d_in[0];
    const float* mask  = (const float*)d_in[1];
    const float* query = (const float*)d_in[2];
    const float* scale = (const float*)d_in[5];
    float* out = (float*)d_out;

    dim3 grid(NQ_ / BN_, BS_);   // (16, 8) = 128 workgroups
    dim3 block(256);             // 8 wave32 waves
    protohead_wmma_kernel<<<grid, block, 0, stream>>>(proto, mask, query, scale, out);
}